// ControlNet_20624432955405
// MI455X (gfx1250) — compile-verified
//
#include <hip/hip_runtime.h>
#include <hip/hip_bf16.h>

// ControlNet scatter-add: y[b, out_idx[k]] = sum(4*x[b,k']) + bias, B=2048, NM=16384.
// Bandwidth-bound (HBM floor ~11us at 23.3 TB/s). Strategy: one workgroup per batch
// row; 64KB LDS accumulator seeded with bias via async global->LDS copy (ASYNCcnt),
// ds_add_f32 LDS atomics for the scatter, async LDS->global writeback, coalesced b128 IO.

#define NM_ELEMS   16384
#define THREADS    256
#define V4_PER_THR (NM_ELEMS / (THREADS * 4))   // 16 float4 chunks per thread

typedef float float4_ __attribute__((ext_vector_type(4)));
typedef int   int4_   __attribute__((ext_vector_type(4)));
typedef int   v4i     __attribute__((vector_size(16)));   // matches builtin param type

#define AS_GLOBAL __attribute__((address_space(1)))
#define AS_LDS    __attribute__((address_space(3)))

__global__ __launch_bounds__(THREADS) void controlnet_scatter_kernel(
    const float* __restrict__ x,
    const float* __restrict__ bias,
    const int*   __restrict__ out_idx,
    float*       __restrict__ y)
{
    __shared__ float acc[NM_ELEMS];   // 64 KB accumulator (WGP has 320 KB LDS)

    const int b = blockIdx.x;
    const int t = threadIdx.x;

    const float4_* bias4 = (const float4_*)bias;

    // ---- Stage 1: seed LDS accumulator with bias (async global->LDS) ------
#if defined(__gfx1250__) && __has_builtin(__builtin_amdgcn_global_load_async_to_lds_b128)
    #pragma unroll
    for (int i = 0; i < V4_PER_THR; ++i) {
        const int i4 = t + i * THREADS;                 // float4 index into row
        __builtin_amdgcn_global_load_async_to_lds_b128(
            (AS_GLOBAL v4i*)(bias4 + i4),
            (AS_LDS    v4i*)(&acc[i4 * 4]),
            /*offset=*/0, /*cpol=*/0);
    }
    #if __has_builtin(__builtin_amdgcn_s_wait_asynccnt)
    __builtin_amdgcn_s_wait_asynccnt(0);
    #else
    asm volatile("s_wait_asynccnt 0" ::: "memory");
    #endif
#else
    // Fallback: plain global load + ds_store
    #pragma unroll
    for (int i = 0; i < V4_PER_THR; ++i) {
        const int i4 = t + i * THREADS;
        ((float4_*)acc)[i4] = bias4[i4];
    }
#endif
    __syncthreads();

    // ---- Stage 2: scatter 4*x into LDS with ds_add_f32 atomics ------------
    const float4_* x4   = (const float4_*)(x + (size_t)b * NM_ELEMS);
    const int4_*   idx4 = (const int4_*)out_idx;        // shared across blocks, L2-resident

    #pragma unroll 4
    for (int i = 0; i < V4_PER_THR; ++i) {
        const int i4 = t + i * THREADS;
        const float4_ xv = x4[i4];                      // coalesced global_load_b128
        const int4_   ov = idx4[i4];
        atomicAdd(&acc[ov.x], 4.0f * xv.x);             // ds_add_f32 (no return)
        atomicAdd(&acc[ov.y], 4.0f * xv.y);
        atomicAdd(&acc[ov.z], 4.0f * xv.z);
        atomicAdd(&acc[ov.w], 4.0f * xv.w);
    }
    __syncthreads();

    // ---- Stage 3: write-back (async LDS->global) --------------------------
    float4_* y4 = (float4_*)(y + (size_t)b * NM_ELEMS);
#if defined(__gfx1250__) && __has_builtin(__builtin_amdgcn_global_store_async_from_lds_b128)
    #pragma unroll
    for (int i = 0; i < V4_PER_THR; ++i) {
        const int i4 = t + i * THREADS;
        __builtin_amdgcn_global_store_async_from_lds_b128(
            (AS_GLOBAL v4i*)(y4 + i4),
            (AS_LDS    v4i*)(&acc[i4 * 4]),
            /*offset=*/0, /*cpol=*/0);
    }
    // Drain ASYNCcnt before the wave ends (S_ENDPGM also implies wait-idle).
    #if __has_builtin(__builtin_amdgcn_s_wait_asynccnt)
    __builtin_amdgcn_s_wait_asynccnt(0);
    #else
    asm volatile("s_wait_asynccnt 0" ::: "memory");
    #endif
#else
    #pragma unroll
    for (int i = 0; i < V4_PER_THR; ++i) {
        const int i4 = t + i * THREADS;
        y4[i4] = ((const float4_*)acc)[i4];
    }
#endif
}

extern "C" void kernel_launch(void* const* d_in, const int* in_sizes, int n_in,
                              void* d_out, int out_size, void* d_ws, size_t ws_size,
                              hipStream_t stream) {
    const float* x       = (const float*)d_in[0];   // [B, 128, 128] f32
    const float* bias    = (const float*)d_in[1];   // [16384] f32
    const int*   out_idx = (const int*)d_in[2];     // [16384] i32
    float*       y       = (float*)d_out;           // [B, 128, 128] f32

    const int batches = in_sizes[0] / NM_ELEMS;     // 2048
    controlnet_scatter_kernel<<<batches, THREADS, 0, stream>>>(x, bias, out_idx, y);
}